// RefinedLayer_60773787238721
// MI455X (gfx1250) — compile-verified
//
#include <hip/hip_runtime.h>
#include <math.h>

#define NN 50000
#define NE 800000
#define DD 64

typedef float v2f __attribute__((ext_vector_type(2)));
typedef float v8f __attribute__((ext_vector_type(8)));

__device__ __forceinline__ float wave_sum32(float v) {
  #pragma unroll
  for (int off = 16; off > 0; off >>= 1) v += __shfl_xor(v, off);
  return v;
}

// ---------------------------------------------------------------------------
// Y[nrows x 64] = X[nrows x 64] @ W[64 x 64] (+ bias), fp32 WMMA 16x16x4.
// One block = 16 rows; 4 waves, one 16x16 column tile each.
// ---------------------------------------------------------------------------
__global__ __launch_bounds__(128)
void gemm64_kernel(const float* __restrict__ X, const float* __restrict__ W,
                   const float* __restrict__ bias, float* __restrict__ Y,
                   int nrows) {
  __shared__ float Xs[16][68];
  const int tid = threadIdx.x;
  const int rowbase = blockIdx.x * 16;
  for (int i = tid; i < 16 * DD; i += 128) {
    int r = i >> 6, c = i & 63;
    int row = rowbase + r;
    Xs[r][c] = (row < nrows) ? X[(size_t)row * DD + c] : 0.0f;
  }
  __syncthreads();

  const int wave = tid >> 5;
  const int lane = tid & 31;
  const int ln   = lane & 15;       // row (A) / col (B,C) within tile
  const int half = lane >> 4;       // 0 or 1
  const int koff = half * 2;
  const int n0   = wave * 16;

  v8f acc = {0.f, 0.f, 0.f, 0.f, 0.f, 0.f, 0.f, 0.f};
  #pragma unroll
  for (int kc = 0; kc < DD; kc += 4) {
    v2f a, b;
    a.x = Xs[ln][kc + koff];
    a.y = Xs[ln][kc + koff + 1];
    b.x = W[(size_t)(kc + koff) * DD + n0 + ln];
    b.y = W[(size_t)(kc + koff + 1) * DD + n0 + ln];
    acc = __builtin_amdgcn_wmma_f32_16x16x4_f32(false, a, false, b,
                                                (short)0, acc, false, false);
  }

  float bv = bias ? bias[n0 + ln] : 0.0f;
  #pragma unroll
  for (int r = 0; r < 8; ++r) {
    int row = rowbase + r + half * 8;
    if (row < nrows) Y[(size_t)row * DD + n0 + ln] = acc[r] + bv;
  }
}

// psi[i] = dot(H[i], f_def_w) + f_def_b[0]
__global__ __launch_bounds__(256)
void psi_kernel(const float* __restrict__ H, const float* __restrict__ fw,
                const float* __restrict__ fb, float* __restrict__ psi) {
  int i = blockIdx.x * 256 + threadIdx.x;
  if (i >= NN) return;
  float s = fb[0];
  #pragma unroll 8
  for (int k = 0; k < DD; ++k) s += H[(size_t)i * DD + k] * fw[k];
  psi[i] = s;
}

// One wave (32 lanes) per edge: attention score + pair score.
__global__ __launch_bounds__(256)
void edge_score_kernel(const int* __restrict__ src, const int* __restrict__ tgt,
                       const float* __restrict__ H,
                       const float* __restrict__ Xatt,
                       const float* __restrict__ Xp,
                       const float* __restrict__ Xpp,
                       float* __restrict__ anum, float* __restrict__ bnum,
                       float* __restrict__ aden, float* __restrict__ bden) {
  int e = blockIdx.x * 8 + (threadIdx.x >> 5);
  if (e >= NE) return;
  int lane = threadIdx.x & 31;
  int s = src[e], t = tgt[e];
  const float* ht  = H    + (size_t)t * DD;
  const float* hs  = H    + (size_t)s * DD;
  const float* xas = Xatt + (size_t)s * DD;
  const float* xps = Xp   + (size_t)s * DD;
  const float* xqt = Xpp  + (size_t)t * DD;
  float p1 = ht[lane] * xas[lane] + ht[lane + 32] * xas[lane + 32];
  float p2 = xps[lane] * xqt[lane] + xps[lane + 32] * xqt[lane + 32];
  float p3 = (lane < 6) ? hs[lane] * ht[lane] : 0.0f;  // s_feat dot
  p1 = wave_sum32(p1);
  p2 = wave_sum32(p2);
  p3 = wave_sum32(p3);
  if (lane == 0) {
    float an = __expf(p1 + p3);
    float bn = __expf(p2);
    anum[e] = an;
    bnum[e] = bn;
    atomicAdd(&aden[t], an);
    atomicAdd(&bden[s], bn);
  }
}

// sum_term[src] += beta_e * exp(-psi[tgt])
__global__ __launch_bounds__(256)
void sumterm_kernel(const int* __restrict__ src, const int* __restrict__ tgt,
                    const float* __restrict__ bnum, const float* __restrict__ bden,
                    const float* __restrict__ psi, float* __restrict__ sterm) {
  int e = blockIdx.x * 256 + threadIdx.x;
  if (e >= NE) return;
  int s = src[e], t = tgt[e];
  float beta = bnum[e] / (bden[s] + 1e-9f);
  atomicAdd(&sterm[s], beta * __expf(-psi[t]));
}

// rho = sigmoid(gamma*(d - delta)), d = -log(sum_term + 1e-8), gamma=1, delta=0.5
__global__ __launch_bounds__(256)
void rho_kernel(const float* __restrict__ sterm, float* __restrict__ rho) {
  int i = blockIdx.x * 256 + threadIdx.x;
  if (i >= NN) return;
  float d = -__logf(sterm[i] + 1e-8f);
  float x = d - 0.5f;
  rho[i] = 1.0f / (1.0f + __expf(-x));
}

// m_att[tgt] += alpha * (1-rho[src]) * Xphi[src]; thread per (edge, feature)
__global__ __launch_bounds__(256)
void msg_kernel(const int* __restrict__ src, const int* __restrict__ tgt,
                const float* __restrict__ Xphi,
                const float* __restrict__ anum, const float* __restrict__ aden,
                const float* __restrict__ rho, float* __restrict__ Matt) {
  int idx = blockIdx.x * 256 + threadIdx.x;   // < NE*64 = 51.2M, fits int
  if (idx >= NE * DD) return;
  int e = idx >> 6;
  int k = idx & 63;
  int s = src[e], t = tgt[e];
  float alpha = anum[e] / (aden[t] + 1e-9f);
  float coef = alpha * (1.0f - rho[s]);
  atomicAdd(&Matt[(size_t)t * DD + k], coef * Xphi[(size_t)s * DD + k]);
}

// ---------------------------------------------------------------------------
// out = LN(relu(H@Wself + b + Matt@WA + b + sfeat@Wstr + b) + H) * g + b
// WMMA for all three GEMMs (Wstr K padded 6->8), LN in LDS.
// ---------------------------------------------------------------------------
__global__ __launch_bounds__(128)
void out_kernel(const float* __restrict__ H, const float* __restrict__ Matt,
                const float* __restrict__ Wself, const float* __restrict__ bself,
                const float* __restrict__ WA, const float* __restrict__ bA,
                const float* __restrict__ Wstr, const float* __restrict__ bstr,
                const float* __restrict__ lng, const float* __restrict__ lnb,
                float* __restrict__ out, int nrows) {
  __shared__ float Hs[16][68];
  __shared__ float Ms[16][68];
  __shared__ float Ss[16][8];
  __shared__ float Os[16][68];
  const int tid = threadIdx.x;
  const int rowbase = blockIdx.x * 16;
  for (int i = tid; i < 16 * DD; i += 128) {
    int r = i >> 6, c = i & 63;
    int row = rowbase + r;
    bool ok = row < nrows;
    Hs[r][c] = ok ? H[(size_t)row * DD + c] : 0.0f;
    Ms[r][c] = ok ? Matt[(size_t)row * DD + c] : 0.0f;
  }
  {
    int r = tid >> 3, c = tid & 7;   // 128 threads == 16*8
    int row = rowbase + r;
    Ss[r][c] = (c < 6 && row < nrows) ? H[(size_t)row * DD + c] : 0.0f;
  }
  __syncthreads();

  const int lane = tid & 31;
  const int ln   = lane & 15;
  const int half = lane >> 4;
  const int koff = half * 2;
  const int n0   = (tid >> 5) * 16;
  const int col  = n0 + ln;

  v8f acc = {0.f, 0.f, 0.f, 0.f, 0.f, 0.f, 0.f, 0.f};
  #pragma unroll
  for (int kc = 0; kc < DD; kc += 4) {             // H @ W_self
    v2f a, b;
    a.x = Hs[ln][kc + koff];       a.y = Hs[ln][kc + koff + 1];
    b.x = Wself[(size_t)(kc + koff) * DD + col];
    b.y = Wself[(size_t)(kc + koff + 1) * DD + col];
    acc = __builtin_amdgcn_wmma_f32_16x16x4_f32(false, a, false, b,
                                                (short)0, acc, false, false);
  }
  #pragma unroll
  for (int kc = 0; kc < DD; kc += 4) {             // Matt @ W_A
    v2f a, b;
    a.x = Ms[ln][kc + koff];       a.y = Ms[ln][kc + koff + 1];
    b.x = WA[(size_t)(kc + koff) * DD + col];
    b.y = WA[(size_t)(kc + koff + 1) * DD + col];
    acc = __builtin_amdgcn_wmma_f32_16x16x4_f32(false, a, false, b,
                                                (short)0, acc, false, false);
  }
  #pragma unroll
  for (int kc = 0; kc < 8; kc += 4) {              // sfeat @ W_str (K=6 padded)
    int r0 = kc + koff, r1 = r0 + 1;
    v2f a, b;
    a.x = Ss[ln][r0];              a.y = Ss[ln][r1];
    b.x = (r0 < 6) ? Wstr[(size_t)r0 * DD + col] : 0.0f;
    b.y = (r1 < 6) ? Wstr[(size_t)r1 * DD + col] : 0.0f;
    acc = __builtin_amdgcn_wmma_f32_16x16x4_f32(false, a, false, b,
                                                (short)0, acc, false, false);
  }

  float bsum = bself[col] + bA[col] + bstr[col];
  #pragma unroll
  for (int r = 0; r < 8; ++r) {
    int rl = r + half * 8;
    float v = acc[r] + bsum;
    v = fmaxf(v, 0.0f) + Hs[rl][col];              // relu + residual
    Os[rl][col] = v;
  }
  __syncthreads();

  if (tid < 16) {
    int row = rowbase + tid;
    if (row < nrows) {
      float mu = 0.0f;
      #pragma unroll 8
      for (int c = 0; c < DD; ++c) mu += Os[tid][c];
      mu *= (1.0f / 64.0f);
      float var = 0.0f;
      #pragma unroll 8
      for (int c = 0; c < DD; ++c) { float d = Os[tid][c] - mu; var += d * d; }
      var *= (1.0f / 64.0f);
      float inv = rsqrtf(var + 1e-5f);
      #pragma unroll 8
      for (int c = 0; c < DD; ++c)
        out[(size_t)row * DD + c] = (Os[tid][c] - mu) * inv * lng[c] + lnb[c];
    }
  }
}

extern "C" void kernel_launch(void* const* d_in, const int* in_sizes, int n_in,
                              void* d_out, int out_size, void* d_ws, size_t ws_size,
                              hipStream_t stream) {
  (void)in_sizes; (void)n_in; (void)out_size; (void)ws_size;
  const int*   ei    = (const int*)d_in[0];
  const int*   src   = ei;
  const int*   tgt   = ei + NE;
  const float* H     = (const float*)d_in[1];
  const float* W_att = (const float*)d_in[2];
  const float* phi_w = (const float*)d_in[3];
  const float* phi_b = (const float*)d_in[4];
  const float* W_p   = (const float*)d_in[5];
  const float* W_pp  = (const float*)d_in[6];
  const float* f_w   = (const float*)d_in[7];
  const float* f_b   = (const float*)d_in[8];
  const float* Wself = (const float*)d_in[9];
  const float* bself = (const float*)d_in[10];
  const float* WA    = (const float*)d_in[11];
  const float* bA    = (const float*)d_in[12];
  const float* Wstr  = (const float*)d_in[13];
  const float* bstr  = (const float*)d_in[14];
  const float* lng   = (const float*)d_in[15];
  const float* lnb   = (const float*)d_in[16];

  float* ws    = (float*)d_ws;
  float* Xatt  = ws;
  float* Xp    = ws + 1ull * NN * DD;
  float* Xpp   = ws + 2ull * NN * DD;
  float* Xphi  = ws + 3ull * NN * DD;
  float* Matt  = ws + 4ull * NN * DD;
  float* psi   = ws + 5ull * NN * DD;
  float* aden  = psi  + NN;
  float* bden  = aden + NN;
  float* sterm = bden + NN;
  float* rho   = sterm + NN;
  float* anum  = rho + NN;
  float* bnum  = anum + NE;

  hipMemsetAsync(Matt, 0, sizeof(float) * (size_t)NN * DD, stream);
  hipMemsetAsync(aden, 0, sizeof(float) * (size_t)3 * NN, stream);  // aden,bden,sterm

  dim3 gb((NN + 15) / 16);
  gemm64_kernel<<<gb, 128, 0, stream>>>(H, W_att, nullptr, Xatt, NN);
  gemm64_kernel<<<gb, 128, 0, stream>>>(H, W_p,   nullptr, Xp,   NN);
  gemm64_kernel<<<gb, 128, 0, stream>>>(H, W_pp,  nullptr, Xpp,  NN);
  gemm64_kernel<<<gb, 128, 0, stream>>>(H, phi_w, phi_b,   Xphi, NN);
  psi_kernel<<<(NN + 255) / 256, 256, 0, stream>>>(H, f_w, f_b, psi);
  edge_score_kernel<<<(NE + 7) / 8, 256, 0, stream>>>(src, tgt, H, Xatt, Xp, Xpp,
                                                      anum, bnum, aden, bden);
  sumterm_kernel<<<(NE + 255) / 256, 256, 0, stream>>>(src, tgt, bnum, bden, psi, sterm);
  rho_kernel<<<(NN + 255) / 256, 256, 0, stream>>>(sterm, rho);
  msg_kernel<<<(NE * DD + 255) / 256, 256, 0, stream>>>(src, tgt, Xphi, anum, aden,
                                                        rho, Matt);
  out_kernel<<<gb, 128, 0, stream>>>(H, Matt, Wself, bself, WA, bA, Wstr, bstr,
                                     lng, lnb, (float*)d_out, NN);
}